// EMACodebook_58428735095072
// MI455X (gfx1250) — compile-verified
//
#include <hip/hip_runtime.h>

// EMACodebook nearest-neighbor quantization for MI455X (gfx1250, wave32).
// score[n,k] = ||w_k||^2 - 2 x_n . w_k  (||x||^2 dropped: constant per row)
// GEMM via v_wmma_f32_16x16x4_f32 (fp32 accumulate - argmin needs precision).

typedef float v2f __attribute__((ext_vector_type(2)));
typedef float v8f __attribute__((ext_vector_type(8)));

#define DDIM    128   // vector dimension (fixed by problem)
#define CHUNK   64    // codebook entries staged in LDS per iteration
#define LSTRIDE 132   // padded floats per LDS row: bank = (4*lane + off) % 64 -> conflict-free b64 loads

// ---------------------------------------------------------------- wnorm ----
__global__ __launch_bounds__(256) void wnorm_k(const float* __restrict__ w,
                                               float* __restrict__ wn, int K) {
  const int lane = threadIdx.x & 31;
  const int wave = threadIdx.x >> 5;
  const int row  = blockIdx.x * 8 + wave;
  if (row >= K) return;
  float4 v = *(const float4*)(w + (size_t)row * DDIM + lane * 4);
  float s = v.x * v.x + v.y * v.y + v.z * v.z + v.w * v.w;
  #pragma unroll
  for (int m = 16; m >= 1; m >>= 1) s += __shfl_xor(s, m, 32);
  if (lane == 0) wn[row] = s;
}

// --------------------------------------------------------------- argmin ----
// Block = 256 threads = 8 waves; each wave owns a 16-row x-tile (block: 128 rows).
__global__ __launch_bounds__(256) void argmin_k(const float* __restrict__ x,
                                                const float* __restrict__ w,
                                                const float* __restrict__ wn,
                                                int* __restrict__ idx_out, int K) {
  __shared__ float lds_w[CHUNK * LSTRIDE];  // 64*132*4 = 33 KB

  const int lane = threadIdx.x & 31;
  const int wave = threadIdx.x >> 5;
  const int half = lane >> 4;   // k-split within WMMA operands
  const int r    = lane & 15;   // row (A) / column (B,C,D) within tile
  const int row0 = (blockIdx.x * 8 + wave) * 16;

  // A operand resident in registers: lane holds x[row0+r][4s+2*half .. +1] for s=0..31
  v2f a[DDIM / 4];
  {
    const float* xr = x + (size_t)(row0 + r) * DDIM + 2 * half;
    #pragma unroll
    for (int s = 0; s < DDIM / 4; ++s) a[s] = *(const v2f*)(xr + 4 * s);
  }

  float best[8];
  int   bidx[8];
  #pragma unroll
  for (int v = 0; v < 8; ++v) { best[v] = 3.4e38f; bidx[v] = 0; }

  for (int cb0 = 0; cb0 < K; cb0 += CHUNK) {
    __syncthreads();  // previous chunk's B reads done before overwrite
    // cooperative stage: CHUNK rows x 32 float4, consecutive threads -> contiguous 128B
    for (int t = threadIdx.x; t < CHUNK * (DDIM / 4); t += 256) {
      const int rr = t >> 5, cc = t & 31;
      *(float4*)(&lds_w[rr * LSTRIDE + cc * 4]) =
          *(const float4*)(w + (size_t)(cb0 + rr) * DDIM + cc * 4);
    }
    __syncthreads();

    for (int jt = 0; jt < CHUNK / 16; ++jt) {          // 16-column tiles
      const int  cbase = cb0 + jt * 16;
      const float* bb  = &lds_w[(jt * 16 + r) * LSTRIDE + 2 * half];

      // Preload full B tile to registers: one LDS burst, single dscnt drain,
      // then WMMAs issue back-to-back (no interleaved s_wait_dscnt).
      v2f b[DDIM / 4];
      #pragma unroll
      for (int s = 0; s < DDIM / 4; ++s) b[s] = *(const v2f*)(bb + 4 * s);

      // Two independent accumulation chains to halve serial WMMA depth.
      v8f acc0 = {};
      v8f acc1 = {};
      #pragma unroll
      for (int s = 0; s < DDIM / 4; s += 2) {
        acc0 = __builtin_amdgcn_wmma_f32_16x16x4_f32(
            false, a[s],     false, b[s],     (short)0, acc0, false, false);
        acc1 = __builtin_amdgcn_wmma_f32_16x16x4_f32(
            false, a[s + 1], false, b[s + 1], (short)0, acc1, false, false);
      }

      const float wnv = wn[cbase + r];                 // column ||w||^2 (lane col = r)
      const int   cix = cbase + r;
      #pragma unroll
      for (int v = 0; v < 8; ++v) {                    // rows M = v (lanes<16), v+8 (lanes>=16)
        const float s2 = fmaf(-2.0f, acc0[v] + acc1[v], wnv);
        if (s2 < best[v]) { best[v] = s2; bidx[v] = cix; }
      }
    }
  }

  // argmin across the 16 lanes of each half (xor masks < 16 stay in-group)
  #pragma unroll
  for (int v = 0; v < 8; ++v) {
    float s = best[v]; int i = bidx[v];
    #pragma unroll
    for (int m = 8; m >= 1; m >>= 1) {
      const float s2 = __shfl_xor(s, m, 32);
      const int   i2 = __shfl_xor(i, m, 32);
      if (s2 < s || (s2 == s && i2 < i)) { s = s2; i = i2; }
    }
    if (r == 0) idx_out[row0 + v + half * 8] = i;      // lanes 0 and 16 write
  }
}

// ------------------------------------------------------------- writeout ----
// One wave per row: gather xq row, stream one-hot p row (float4 stores).
__global__ __launch_bounds__(256) void writeout_k(const float* __restrict__ w,
                                                  const int* __restrict__ idx,
                                                  float* __restrict__ xq,
                                                  float* __restrict__ p,
                                                  int N, int K) {
  const int lane = threadIdx.x & 31;
  const int wave = threadIdx.x >> 5;
  const int row  = blockIdx.x * 8 + wave;
  if (row >= N) return;
  int id = idx[row];
  id = __builtin_amdgcn_readfirstlane(id);

  *(float4*)(xq + (size_t)row * DDIM + lane * 4) =
      *(const float4*)(w + (size_t)id * DDIM + lane * 4);

  float* prow = p + (size_t)row * K;
  for (int k0 = lane * 4; k0 < K; k0 += 128) {
    float4 v;
    v.x = (k0     == id) ? 1.0f : 0.0f;
    v.y = (k0 + 1 == id) ? 1.0f : 0.0f;
    v.z = (k0 + 2 == id) ? 1.0f : 0.0f;
    v.w = (k0 + 3 == id) ? 1.0f : 0.0f;
    *(float4*)(prow + k0) = v;
  }
}

// ---------------------------------------------------------------- launch ---
extern "C" void kernel_launch(void* const* d_in, const int* in_sizes, int n_in,
                              void* d_out, int out_size, void* d_ws, size_t ws_size,
                              hipStream_t stream) {
  const float* x = (const float*)d_in[0];
  const float* w = (const float*)d_in[1];
  const int N = in_sizes[0] / DDIM;   // 32768
  const int K = in_sizes[1] / DDIM;   // 1024

  float* wn  = (float*)d_ws;
  int*   idx = (int*)((char*)d_ws + (((size_t)K * sizeof(float) + 255) & ~(size_t)255));
  float* xq  = (float*)d_out;
  float* p   = xq + (size_t)N * DDIM;

  wnorm_k   <<<(K + 7) / 8, 256, 0, stream>>>(w, wn, K);
  argmin_k  <<<N / 128,     256, 0, stream>>>(x, w, wn, idx, K);
  writeout_k<<<(N + 7) / 8, 256, 0, stream>>>(w, idx, xq, p, N, K);
}